// GraphDataNet_38508676776060
// MI455X (gfx1250) — compile-verified
//
#include <hip/hip_runtime.h>
#include <math.h>

#ifndef __has_builtin
#define __has_builtin(x) 0
#endif

// gfx1250 hardware tanh (v_tanh_f32) when the toolchain exposes it.
__device__ __forceinline__ float dev_tanhf(float x) {
#if __has_builtin(__builtin_amdgcn_tanhf)
    return __builtin_amdgcn_tanhf(x);
#elif __has_builtin(__builtin_amdgcn_tanh_f32)
    return __builtin_amdgcn_tanh_f32(x);
#else
    return tanhf(x);
#endif
}

// Native fp32 global atomic add (global_atomic_add_f32, no return -> STOREcnt).
__device__ __forceinline__ void atomic_add_f32(float* p, float v) {
    __hip_atomic_fetch_add(p, v, __ATOMIC_RELAXED, __HIP_MEMORY_SCOPE_AGENT);
}

static constexpr int BB = 16;  // batch dim fixed by the reference

// ---------------------------------------------------------------------------
// Kernel 1: transpose x (B,N) -> xt (N,16).
// Reads coalesced per-b (consecutive threads = consecutive n), each thread
// writes its own contiguous 64B row via 4x global_store_b128.
// ---------------------------------------------------------------------------
__global__ __launch_bounds__(256)
void k_transpose(const float* __restrict__ x, float* __restrict__ xt, int N) {
    int n = blockIdx.x * blockDim.x + threadIdx.x;
    if (n >= N) return;
    float4 v[4];
    float* f = reinterpret_cast<float*>(v);
#pragma unroll
    for (int b = 0; b < BB; ++b)
        f[b] = x[(size_t)b * N + n];
    float4* dst = reinterpret_cast<float4*>(xt + (size_t)n * BB);
#pragma unroll
    for (int i = 0; i < 4; ++i)
        dst[i] = v[i];
}

// ---------------------------------------------------------------------------
// Kernel 2: per-edge gather -> alpha-blended tanh -> scatter-add.
// One thread per edge: streamed index/param loads (coalesced), one contiguous
// 64B gather (4x global_load_b128 from L2-resident xt), 16 contiguous
// global_atomic_add_f32 into the (N,16) accumulator.
// ---------------------------------------------------------------------------
__global__ __launch_bounds__(256)
void k_edges(const float* __restrict__ xt,
             const int*   __restrict__ src_idx,
             const int*   __restrict__ dst_idx,
             const float* __restrict__ e_alpha,
             const float* __restrict__ e_w,
             const float* __restrict__ e_b,
             float* __restrict__ yacc, int E) {
    int e = blockIdx.x * blockDim.x + threadIdx.x;
    if (e >= E) return;

    // gfx1250 global_prefetch_b8 for the streamed edge arrays (speculative,
    // OOB-safe: translation failures are silently dropped).
    __builtin_prefetch(&src_idx[e + 4096], 0, 1);
    __builtin_prefetch(&e_w[e + 4096], 0, 1);

    const int   s = src_idx[e];
    const int   d = dst_idx[e];
    const float a = e_alpha[e];
    const float w = e_w[e];
    const float b = e_b[e];

    const float4* xs = reinterpret_cast<const float4*>(xt + (size_t)s * BB);
    float*        yd = yacc + (size_t)d * BB;

#pragma unroll
    for (int i = 0; i < 4; ++i) {
        float4 v = xs[i];
        float l0 = fmaf(w, v.x, b);
        float l1 = fmaf(w, v.y, b);
        float l2 = fmaf(w, v.z, b);
        float l3 = fmaf(w, v.w, b);
        // (1-a)*lin + a*tanh(lin) == lin + a*(tanh(lin)-lin)
        float m0 = fmaf(a, dev_tanhf(l0) - l0, l0);
        float m1 = fmaf(a, dev_tanhf(l1) - l1, l1);
        float m2 = fmaf(a, dev_tanhf(l2) - l2, l2);
        float m3 = fmaf(a, dev_tanhf(l3) - l3, l3);
        atomic_add_f32(&yd[i * 4 + 0], m0);
        atomic_add_f32(&yd[i * 4 + 1], m1);
        atomic_add_f32(&yd[i * 4 + 2], m2);
        atomic_add_f32(&yd[i * 4 + 3], m3);
    }
}

// ---------------------------------------------------------------------------
// Kernel 3: node nonlinearity + transpose back to (B,N).
// Per-thread contiguous 64B read of the accumulator (4x global_load_b128);
// per-b writes are coalesced across consecutive threads.
// ---------------------------------------------------------------------------
__global__ __launch_bounds__(256)
void k_nodes(const float* __restrict__ yacc,
             const float* __restrict__ n_alpha,
             const float* __restrict__ n_w,
             const float* __restrict__ n_b,
             float* __restrict__ out, int N) {
    int n = blockIdx.x * blockDim.x + threadIdx.x;
    if (n >= N) return;

    const float a = n_alpha[n];
    const float w = n_w[n];
    const float b = n_b[n];

    const float4* ys = reinterpret_cast<const float4*>(yacc + (size_t)n * BB);
    float v[BB];
#pragma unroll
    for (int i = 0; i < 4; ++i) {
        float4 q = ys[i];
        v[i * 4 + 0] = q.x; v[i * 4 + 1] = q.y;
        v[i * 4 + 2] = q.z; v[i * 4 + 3] = q.w;
    }
#pragma unroll
    for (int bq = 0; bq < BB; ++bq) {
        float lin = fmaf(w, v[bq], b);
        out[(size_t)bq * N + n] = fmaf(a, dev_tanhf(lin) - lin, lin);
    }
}

// ---------------------------------------------------------------------------
extern "C" void kernel_launch(void* const* d_in, const int* in_sizes, int n_in,
                              void* d_out, int out_size, void* d_ws, size_t ws_size,
                              hipStream_t stream) {
    const float* x       = (const float*)d_in[0];
    const int*   src_idx = (const int*)  d_in[1];
    const int*   dst_idx = (const int*)  d_in[2];
    const float* e_alpha = (const float*)d_in[3];
    const float* e_w     = (const float*)d_in[4];
    const float* e_b     = (const float*)d_in[5];
    const float* n_alpha = (const float*)d_in[6];
    const float* n_w     = (const float*)d_in[7];
    const float* n_b     = (const float*)d_in[8];

    const int E = in_sizes[1];       // edges
    const int N = in_sizes[6];       // nodes (node_alpha length)

    float* xt   = (float*)d_ws;                    // (N,16) transposed x
    float* yacc = xt + (size_t)N * BB;             // (N,16) accumulator

    // Accumulator must be zeroed every call (graph-capture legal memset node).
    hipMemsetAsync(yacc, 0, (size_t)N * BB * sizeof(float), stream);

    const int blk = 256;
    k_transpose<<<(N + blk - 1) / blk, blk, 0, stream>>>(x, xt, N);
    k_edges    <<<(E + blk - 1) / blk, blk, 0, stream>>>(xt, src_idx, dst_idx,
                                                         e_alpha, e_w, e_b, yacc, E);
    k_nodes    <<<(N + blk - 1) / blk, blk, 0, stream>>>(yacc, n_alpha, n_w, n_b,
                                                         (float*)d_out, N);
}